// RSSM_38328288149704
// MI455X (gfx1250) — compile-verified
//
#include <hip/hip_runtime.h>
#include <hip/hip_bf16.h>

// ---------------------------------------------------------------------------
// RSSM cell step for MI455X (gfx1250), fp32 via V_WMMA_F32_16X16X4_F32.
// Memory-bound: fuse the four [4096x4096] GEMMs + SSM elementwise update;
// stage all GEMM tiles with GLOBAL_LOAD_ASYNC_TO_LDS_B128 (ASYNCcnt) and
// double-buffer LDS so global fetch latency overlaps the WMMA stream.
// ---------------------------------------------------------------------------

typedef __attribute__((ext_vector_type(2))) float v2f;
typedef __attribute__((ext_vector_type(8))) float v8f;

#define BATCH   8192
#define STOCH   1024
#define ACT     6
#define HIDDEN  1024
#define DETER   4096
#define KPAD    1056          // STOCH+ACT=1030 padded up to a multiple of 32
#define EPS     1e-4f

#define TB_M    128           // block tile M
#define TB_N    64            // block tile N
#define KB_G    32            // K slab, generic GEMM
#define LDA_G   (KB_G + 4)    // 36 floats: 144B stride -> 16B aligned, banks spread
#define KB_F    16            // K slab, fused 4-weight GEMM (LDS budget)
#define LDA_F   (KB_F + 4)    // 20 floats: 80B stride -> 16B aligned, banks spread

// ---------------------------------------------------------------------------
// CDNA5 async global->LDS copy (16B per lane), tracked by ASYNCcnt.

__device__ __forceinline__ unsigned lds_off32(const void* p) {
    typedef __attribute__((address_space(3))) const void lds_cv;
    return (unsigned)(unsigned long long)(lds_cv*)p;    // LDS byte offset
}

__device__ __forceinline__ void async_cp16(const void* g, void* l) {
    asm volatile("global_load_async_to_lds_b128 %0, %1, off"
                 :: "v"(lds_off32(l)), "v"((unsigned long long)g)
                 : "memory");
}

// Wait until ASYNCcnt <= n (async loads complete in order).
__device__ __forceinline__ void wait_async(int n) {
    if (n == 0) asm volatile("s_wait_asynccnt 0x0" ::: "memory");
    else        asm volatile("s_wait_asynccnt 0x6" ::: "memory");
}

// ---------------------------------------------------------------------------

__device__ __forceinline__ float softplus_f(float t) {
    return t > 20.0f ? t : log1pf(expf(t));
}
__device__ __forceinline__ float sigmoid_f(float t) {
    return 1.0f / (1.0f + expf(-t));
}

__device__ __forceinline__ float block_reduce_sum(float v, float* red) {
    const int tid = threadIdx.x;
    #pragma unroll
    for (int off = 16; off > 0; off >>= 1) v += __shfl_xor(v, off, 32);
    if ((tid & 31) == 0) red[tid >> 5] = v;
    __syncthreads();
    if (tid == 0) {
        float total = 0.0f;
        #pragma unroll
        for (int i = 0; i < 8; ++i) total += red[i];
        red[0] = total;
    }
    __syncthreads();
    return red[0];
}

// ---------------------------------------------------------------------------
// Prep: tokp[row, 0:1024]=stoch, [1024:1030]=action/clip(|a|,1), rest 0.
__global__ __launch_bounds__(256)
void rssm_prep_tok(const float* __restrict__ stoch, const float* __restrict__ action,
                   float* __restrict__ tokp) {
    const int row = blockIdx.x;
    for (int i = threadIdx.x; i < KPAD; i += 256) {
        float v = 0.0f;
        if (i < STOCH) {
            v = stoch[(size_t)row * STOCH + i];
        } else if (i < STOCH + ACT) {
            float a = action[(size_t)row * ACT + (i - STOCH)];
            v = a / fmaxf(fabsf(a), 1.0f);
        }
        tokp[(size_t)row * KPAD + i] = v;
    }
}

// Prep: zero-pad W_t1 [1024,1030] -> [1024,1056]
__global__ __launch_bounds__(256)
void rssm_prep_w1(const float* __restrict__ W, float* __restrict__ Wp) {
    const int idx = blockIdx.x * 256 + threadIdx.x;
    if (idx < HIDDEN * KPAD) {
        const int r = idx / KPAD, c = idx - r * KPAD;
        Wp[idx] = (c < STOCH + ACT) ? W[(size_t)r * (STOCH + ACT) + c] : 0.0f;
    }
}

// ---------------------------------------------------------------------------
// Generic  C[M,N] = X[M,K] @ W[N,K]^T + bias[N]   (K % 32 == 0, M%128, N%64)
// 256 threads = 8 waves; wave grid 4(M) x 2(N); each wave: 32x32 output.
// Double-buffered LDS, async global->LDS staging (6 x b128 per thread/stage).
__global__ __launch_bounds__(256)
void rssm_gemm_bias(const float* __restrict__ X, const float* __restrict__ W,
                    const float* __restrict__ bias, float* __restrict__ C,
                    int M, int N, int K) {
    __shared__ float Xs[2][TB_M * LDA_G];
    __shared__ float Ws[2][TB_N * LDA_G];

    const int tid  = threadIdx.x;
    const int lane = tid & 31;
    const int wave = tid >> 5;
    const int wm   = wave >> 1;          // 0..3
    const int wn   = wave & 1;           // 0..1
    const int r    = lane & 15;          // row/col within 16x16 tile
    const int kh   = lane >> 4;          // lane-half selects K pair (ISA 7.12.2)

    const int m_blk = blockIdx.y * TB_M;
    const int n_blk = blockIdx.x * TB_N;

    v8f acc[2][2];
    #pragma unroll
    for (int tn = 0; tn < 2; ++tn) {
        const float bv = bias[n_blk + wn * 32 + tn * 16 + r];
        #pragma unroll
        for (int tm = 0; tm < 2; ++tm)
            #pragma unroll
            for (int v = 0; v < 8; ++v) acc[tm][tn][v] = bv;
    }

    auto prefetch = [&](int k0, int buf) {
        #pragma unroll
        for (int i = tid; i < TB_M * (KB_G / 4); i += 256) {   // 4 per thread
            const int row = i >> 3, c4 = i & 7;
            async_cp16(&X[(size_t)(m_blk + row) * K + k0 + c4 * 4],
                       &Xs[buf][row * LDA_G + c4 * 4]);
        }
        #pragma unroll
        for (int i = tid; i < TB_N * (KB_G / 4); i += 256) {   // 2 per thread
            const int row = i >> 3, c4 = i & 7;
            async_cp16(&W[(size_t)(n_blk + row) * K + k0 + c4 * 4],
                       &Ws[buf][row * LDA_G + c4 * 4]);
        }
    };

    const int nslab = K / KB_G;
    prefetch(0, 0);
    for (int s = 0; s < nslab; ++s) {
        const int cur = s & 1;
        const bool more = (s + 1) < nslab;
        if (more) prefetch((s + 1) * KB_G, cur ^ 1);
        wait_async(more ? 6 : 0);        // current slab landed; next still in flight
        __syncthreads();

        #pragma unroll
        for (int kk = 0; kk < KB_G; kk += 4) {
            const int ko = kk + kh * 2;
            v2f a0 = *(const v2f*)&Xs[cur][(wm * 32 +  0 + r) * LDA_G + ko];
            v2f a1 = *(const v2f*)&Xs[cur][(wm * 32 + 16 + r) * LDA_G + ko];
            v2f b0 = *(const v2f*)&Ws[cur][(wn * 32 +  0 + r) * LDA_G + ko];
            v2f b1 = *(const v2f*)&Ws[cur][(wn * 32 + 16 + r) * LDA_G + ko];
            acc[0][0] = __builtin_amdgcn_wmma_f32_16x16x4_f32(false, a0, false, b0, (short)0, acc[0][0], false, false);
            acc[0][1] = __builtin_amdgcn_wmma_f32_16x16x4_f32(false, a0, false, b1, (short)0, acc[0][1], false, false);
            acc[1][0] = __builtin_amdgcn_wmma_f32_16x16x4_f32(false, a1, false, b0, (short)0, acc[1][0], false, false);
            acc[1][1] = __builtin_amdgcn_wmma_f32_16x16x4_f32(false, a1, false, b1, (short)0, acc[1][1], false, false);
        }
        __syncthreads();                 // buffer cur reusable for prefetch s+2
    }

    #pragma unroll
    for (int tm = 0; tm < 2; ++tm)
        #pragma unroll
        for (int tn = 0; tn < 2; ++tn)
            #pragma unroll
            for (int v = 0; v < 8; ++v) {
                const int row = m_blk + wm * 32 + tm * 16 + v + 8 * kh;
                const int col = n_blk + wn * 32 + tn * 16 + r;
                C[(size_t)row * N + col] = acc[tm][tn][v];
            }
}

// ---------------------------------------------------------------------------
// h = silu(rmsnorm(t1, g_t)) in place, rows of HIDDEN=1024, 256 thr x 4 elems
__global__ __launch_bounds__(256)
void rssm_rmsnorm_silu(float* __restrict__ T, const float* __restrict__ g) {
    __shared__ float red[8];
    const int row = blockIdx.x;
    float* rowp = T + (size_t)row * HIDDEN;
    const int tid = threadIdx.x;
    float vals[4];
    float ss = 0.0f;
    #pragma unroll
    for (int j = 0; j < 4; ++j) {
        float t = rowp[tid + j * 256];
        vals[j] = t;
        ss += t * t;
    }
    const float total = block_reduce_sum(ss, red);
    const float rinv = rsqrtf(total * (1.0f / HIDDEN) + EPS);
    #pragma unroll
    for (int j = 0; j < 4; ++j) {
        const int c = tid + j * 256;
        const float v = vals[j] * rinv * g[c];
        rowp[c] = v * sigmoid_f(v);                 // SiLU
    }
}

// out = rmsnorm(out, g_norm) in place, rows of DETER=4096, 256 thr x 4 float4
__global__ __launch_bounds__(256)
void rssm_rmsnorm_final(float* __restrict__ O, const float* __restrict__ g) {
    __shared__ float red[8];
    const int row = blockIdx.x;
    float* rowp = O + (size_t)row * DETER;
    const int tid = threadIdx.x;
    float4 vals[4];
    float ss = 0.0f;
    #pragma unroll
    for (int j = 0; j < 4; ++j) {
        float4 t = *(const float4*)&rowp[(tid + j * 256) * 4];
        vals[j] = t;
        ss += t.x * t.x + t.y * t.y + t.z * t.z + t.w * t.w;
    }
    const float total = block_reduce_sum(ss, red);
    const float rinv = rsqrtf(total * (1.0f / DETER) + EPS);
    #pragma unroll
    for (int j = 0; j < 4; ++j) {
        const int c = (tid + j * 256) * 4;
        float4 t = vals[j];
        t.x *= rinv * g[c + 0];
        t.y *= rinv * g[c + 1];
        t.z *= rinv * g[c + 2];
        t.w *= rinv * g[c + 3];
        *(float4*)&rowp[c] = t;
    }
}

// ---------------------------------------------------------------------------
// Fused: four GEMMs  x @ {W_dt,W_b,W_c,W_z}^T + bias  followed by the
// position-wise SSM update. Writes pre-norm output directly into d_out.
// Double-buffered LDS (KB=16), async staging (6 x b128 per thread/stage).
__global__ __launch_bounds__(256)
void rssm_ssm_fused(const float* __restrict__ X,      // [B, 4096]
                    const float* __restrict__ W_dt, const float* __restrict__ W_b,
                    const float* __restrict__ W_c,  const float* __restrict__ W_z,
                    const float* __restrict__ b_dt, const float* __restrict__ b_b,
                    const float* __restrict__ b_c,  const float* __restrict__ b_z,
                    const float* __restrict__ deter, const float* __restrict__ a_base,
                    const float* __restrict__ skip,  float* __restrict__ out) {
    __shared__ float Xs[2][TB_M * LDA_F];
    __shared__ float Ws[2][4][TB_N * LDA_F];

    const int tid  = threadIdx.x;
    const int lane = tid & 31;
    const int wave = tid >> 5;
    const int wm   = wave >> 1;
    const int wn   = wave & 1;
    const int r    = lane & 15;
    const int kh   = lane >> 4;

    const int m_blk = blockIdx.y * TB_M;
    const int n_blk = blockIdx.x * TB_N;

    const float* Wp[4] = { W_dt, W_b, W_c, W_z };
    const float* bp[4] = { b_dt, b_b, b_c, b_z };

    v8f acc[4][2][2];                                 // [weight][tm][tn]
    #pragma unroll
    for (int q = 0; q < 4; ++q)
        #pragma unroll
        for (int tn = 0; tn < 2; ++tn) {
            const float bv = bp[q][n_blk + wn * 32 + tn * 16 + r];
            #pragma unroll
            for (int tm = 0; tm < 2; ++tm)
                #pragma unroll
                for (int v = 0; v < 8; ++v) acc[q][tm][tn][v] = bv;
        }

    auto prefetch = [&](int k0, int buf) {
        #pragma unroll
        for (int i = tid; i < TB_M * (KB_F / 4); i += 256) {       // 2 per thread
            const int row = i >> 2, c4 = i & 3;
            async_cp16(&X[(size_t)(m_blk + row) * DETER + k0 + c4 * 4],
                       &Xs[buf][row * LDA_F + c4 * 4]);
        }
        #pragma unroll
        for (int i = tid; i < 4 * TB_N * (KB_F / 4); i += 256) {   // 4 per thread
            const int q = i >> 8, j = i & 255;
            const int row = j >> 2, c4 = j & 3;
            async_cp16(&Wp[q][(size_t)(n_blk + row) * DETER + k0 + c4 * 4],
                       &Ws[buf][q][row * LDA_F + c4 * 4]);
        }
    };

    const int nslab = DETER / KB_F;
    prefetch(0, 0);
    for (int s = 0; s < nslab; ++s) {
        const int cur = s & 1;
        const bool more = (s + 1) < nslab;
        if (more) prefetch((s + 1) * KB_F, cur ^ 1);
        wait_async(more ? 6 : 0);
        __syncthreads();

        #pragma unroll
        for (int kk = 0; kk < KB_F; kk += 4) {
            const int ko = kk + kh * 2;
            v2f a0 = *(const v2f*)&Xs[cur][(wm * 32 +  0 + r) * LDA_F + ko];
            v2f a1 = *(const v2f*)&Xs[cur][(wm * 32 + 16 + r) * LDA_F + ko];
            #pragma unroll
            for (int q = 0; q < 4; ++q) {
                v2f b0 = *(const v2f*)&Ws[cur][q][(wn * 32 +  0 + r) * LDA_F + ko];
                v2f b1 = *(const v2f*)&Ws[cur][q][(wn * 32 + 16 + r) * LDA_F + ko];
                acc[q][0][0] = __builtin_amdgcn_wmma_f32_16x16x4_f32(false, a0, false, b0, (short)0, acc[q][0][0], false, false);
                acc[q][0][1] = __builtin_amdgcn_wmma_f32_16x16x4_f32(false, a0, false, b1, (short)0, acc[q][0][1], false, false);
                acc[q][1][0] = __builtin_amdgcn_wmma_f32_16x16x4_f32(false, a1, false, b0, (short)0, acc[q][1][0], false, false);
                acc[q][1][1] = __builtin_amdgcn_wmma_f32_16x16x4_f32(false, a1, false, b1, (short)0, acc[q][1][1], false, false);
            }
        }
        __syncthreads();
    }

    // Position-wise SSM epilogue on the accumulators.
    #pragma unroll
    for (int tn = 0; tn < 2; ++tn) {
        const int col = n_blk + wn * 32 + tn * 16 + r;
        const float a_col = -softplus_f(a_base[col]);
        const float sk    = skip[col];
        #pragma unroll
        for (int tm = 0; tm < 2; ++tm)
            #pragma unroll
            for (int v = 0; v < 8; ++v) {
                const int row = m_blk + wm * 32 + tm * 16 + v + 8 * kh;
                const size_t idx = (size_t)row * DETER + col;
                const float xv  = X[idx];
                const float det = deter[idx];
                const float dt  = softplus_f(acc[0][tm][tn][v]) + 1e-4f;
                const float bb  = tanhf(acc[1][tm][tn][v]);
                const float cc  = acc[2][tm][tn][v];
                const float zz  = sigmoid_f(acc[3][tm][tn][v]);
                const float st  = expf(a_col * dt) * det + dt * bb * xv;
                const float y   = cc * st + sk * xv;
                out[idx] = zz * y + (1.0f - zz) * det;
            }
    }
}

// ---------------------------------------------------------------------------

extern "C" void kernel_launch(void* const* d_in, const int* in_sizes, int n_in,
                              void* d_out, int out_size, void* d_ws, size_t ws_size,
                              hipStream_t stream) {
    (void)in_sizes; (void)n_in; (void)out_size; (void)ws_size;

    const float* stoch  = (const float*)d_in[0];
    const float* deter  = (const float*)d_in[1];
    const float* action = (const float*)d_in[2];
    const float* W_t1   = (const float*)d_in[3];
    const float* b_t1   = (const float*)d_in[4];
    const float* g_t    = (const float*)d_in[5];
    const float* W_t2   = (const float*)d_in[6];
    const float* b_t2   = (const float*)d_in[7];
    const float* a_base = (const float*)d_in[8];
    const float* W_dt   = (const float*)d_in[9];
    const float* b_dt   = (const float*)d_in[10];
    const float* W_b    = (const float*)d_in[11];
    const float* b_b    = (const float*)d_in[12];
    const float* W_c    = (const float*)d_in[13];
    const float* b_c    = (const float*)d_in[14];
    const float* W_z    = (const float*)d_in[15];
    const float* b_z    = (const float*)d_in[16];
    const float* skip   = (const float*)d_in[17];
    const float* g_norm = (const float*)d_in[18];
    float* out = (float*)d_out;

    // Workspace layout (floats): tokp | Wt1p | t1(h) | x   (~207 MB)
    float* ws   = (float*)d_ws;
    float* tokp = ws;
    float* Wt1p = tokp + (size_t)BATCH * KPAD;
    float* t1   = Wt1p + (size_t)HIDDEN * KPAD;
    float* x    = t1   + (size_t)BATCH * HIDDEN;

    // 1) token prep + W_t1 padding
    rssm_prep_tok<<<BATCH, 256, 0, stream>>>(stoch, action, tokp);
    rssm_prep_w1<<<(HIDDEN * KPAD + 255) / 256, 256, 0, stream>>>(W_t1, Wt1p);

    // 2) t1 = tokp @ Wt1p^T + b_t1     [8192,1056]x[1056,1024]
    rssm_gemm_bias<<<dim3(HIDDEN / TB_N, BATCH / TB_M), 256, 0, stream>>>(
        tokp, Wt1p, b_t1, t1, BATCH, HIDDEN, KPAD);

    // 3) h = silu(rmsnorm(t1, g_t))  (in place)
    rssm_rmsnorm_silu<<<BATCH, 256, 0, stream>>>(t1, g_t);

    // 4) x = h @ W_t2^T + b_t2         [8192,1024]x[1024,4096]
    rssm_gemm_bias<<<dim3(DETER / TB_N, BATCH / TB_M), 256, 0, stream>>>(
        t1, W_t2, b_t2, x, BATCH, DETER, HIDDEN);

    // 5) fused 4x GEMM + SSM elementwise -> pre-norm output in d_out
    rssm_ssm_fused<<<dim3(DETER / TB_N, BATCH / TB_M), 256, 0, stream>>>(
        x, W_dt, W_b, W_c, W_z, b_dt, b_b, b_c, b_z, deter, a_base, skip, out);

    // 6) final rmsnorm in place
    rssm_rmsnorm_final<<<BATCH, 256, 0, stream>>>(out, g_norm);
}